// My_Attention_88639535055384
// MI455X (gfx1250) — compile-verified
//
#include <hip/hip_runtime.h>

// CDNA5 (gfx1250) wave32 WMMA implementation using V_WMMA_F32_16X16X4_F32,
// with async global->LDS staging (ASYNCcnt) for the weighted-sum GEMM.
typedef float v2f __attribute__((ext_vector_type(2)));
typedef float v8f __attribute__((ext_vector_type(8)));
typedef int   v4i_vs __attribute__((vector_size(16)));  // matches builtin param

#define SEQ 80
#define DIM 256
#define T_REAL 11
#define T_PAD 16

#if defined(__gfx1250__) && \
    __has_builtin(__builtin_amdgcn_global_load_async_to_lds_b128) && \
    __has_builtin(__builtin_amdgcn_s_wait_asynccnt)
#define USE_ASYNC_LDS 1
#else
#define USE_ASYNC_LDS 0
#endif

// Copy 16 rows x 256 cols of f32 (16 KB) global -> LDS with a 128-thread
// block: 8 x b128 per thread.  Async path uses GLOBAL_LOAD_ASYNC_TO_LDS_B128.
__device__ __forceinline__ void stage_chunk(const float* __restrict__ src,
                                            float* dst_lds, int tid) {
#if USE_ASYNC_LDS
  #pragma unroll
  for (int i = 0; i < 8; ++i) {
    const int off = (tid + i * 128) * 4;  // float index, 16B aligned
    __builtin_amdgcn_global_load_async_to_lds_b128(
        (v4i_vs*)(src + off), (v4i_vs*)(dst_lds + off), 0, 0);
  }
#else
  #pragma unroll
  for (int i = 0; i < 8; ++i) {
    const int off = (tid + i * 128) * 4;
    *(float4*)(dst_lds + off) = *(const float4*)(src + off);
  }
#endif
}

__device__ __forceinline__ void wait_async_copies() {
#if USE_ASYNC_LDS
  __builtin_amdgcn_s_wait_asynccnt(0);
#endif
}

// ---------------------------------------------------------------------------
// Kernel 1: scores[r][t] = sum_k x2[r][k] * kernel[k][t]   (t padded to 16)
// One wave per 16-row M tile. 64 K-steps of wmma_f32_16x16x4_f32.
//
// A fragment (16x4 f32): lane L<16 -> row M=L, (K0,K1); lane L>=16 -> (K2,K3).
// B fragment (4x16 f32): lane L<16 -> col N=L, (K0,K1); lane L>=16 -> (K2,K3).
// D (16x16 f32, 8 VGPRs): vgpr v, lane L<16 -> D[v][L]; L>=16 -> D[v+8][L-16].
// ---------------------------------------------------------------------------
__global__ void scores_kernel(const float* __restrict__ x,
                              const float* __restrict__ kern,
                              float* __restrict__ sc) {
  __shared__ float kshT[T_PAD * DIM]; // transposed: kshT[t*256 + k]
  const int tid = threadIdx.x;
  for (int idx = tid; idx < T_PAD * DIM; idx += blockDim.x) {
    const int t = idx >> 8;   // 0..15
    const int k = idx & 255;  // 0..255
    kshT[idx] = (t < T_REAL) ? kern[k * T_REAL + t] : 0.0f;
  }
  __syncthreads();

  const int wave = tid >> 5;
  const int lane = tid & 31;
  const int lm = lane & 15;   // M row (A) / N col (B,D)
  const int lh = lane >> 4;   // 0: K pair {0,1}; 1: K pair {2,3}

  const int tile = blockIdx.x * 4 + wave;  // 0..1279
  const int r0 = tile * 16;

  const float* arow = x + (size_t)(r0 + lm) * DIM + 2 * lh;   // 8B aligned
  const float* bcol = kshT + lm * DIM + 2 * lh;               // 8B aligned

  v8f acc = {};
  #pragma unroll 4
  for (int kk = 0; kk < DIM / 4; ++kk) {
    v2f a = *(const v2f*)(arow + kk * 4);
    v2f b = *(const v2f*)(bcol + kk * 4);
    acc = __builtin_amdgcn_wmma_f32_16x16x4_f32(false, a, false, b,
                                                (short)0, acc, false, false);
  }

  // Store D row-major: sc[r][t]; lanes 0-15 / 16-31 each write 64B segments.
  #pragma unroll
  for (int v = 0; v < 8; ++v) {
    const int row = r0 + v + 8 * lh;
    sc[(size_t)row * T_PAD + lm] = acc[v];
  }
}

// ---------------------------------------------------------------------------
// Kernel 2: softmax over groups {r : r mod B == j} (80 members, per topic t).
// Writes w[t][r] = exp(sc[r][t]-max)/sum.  Rows t=11..15 zero-filled so the
// padded WMMA A-fragments of kernel 3 read zeros.
// grid: 16 blocks (t), block: B threads (j).
// ---------------------------------------------------------------------------
__global__ void softmax_kernel(const float* __restrict__ sc,
                               float* __restrict__ w, int B, int rows) {
  const int t = blockIdx.x;   // 0..15
  const int j = threadIdx.x;  // 0..B-1

  if (t >= T_REAL) {
    for (int i = 0; i < SEQ; ++i) w[(size_t)t * rows + j + B * i] = 0.0f;
    return;
  }

  float mx = -3.0e38f;
  for (int i = 0; i < SEQ; ++i) {
    const float s = sc[(size_t)(j + B * i) * T_PAD + t];
    mx = fmaxf(mx, s);
  }
  float sum = 0.0f;
  for (int i = 0; i < SEQ; ++i) {
    const float s = sc[(size_t)(j + B * i) * T_PAD + t];
    sum += __expf(s - mx);
  }
  const float inv = 1.0f / sum;
  for (int i = 0; i < SEQ; ++i) {
    const int r = j + B * i;
    const float s = sc[(size_t)r * T_PAD + t];
    w[(size_t)t * rows + r] = __expf(s - mx) * inv;  // coalesced across lanes
  }
}

// ---------------------------------------------------------------------------
// Kernel 3: v[t][b][d] = sum_s w[t][b*80+s] * x[b][s][d]
// One block (128 threads = 4 waves) per batch b.  The 16x80 weight tile is
// staged once in LDS; x[b] (80x256) is staged in 5 chunks of 16 K-rows with
// double-buffered async global->LDS copies overlapping the WMMA work.
// Each wave owns 4 N-tiles; x is read from global exactly once.
// ---------------------------------------------------------------------------
__global__ void weighted_sum_kernel(const float* __restrict__ x,
                                    const float* __restrict__ w,
                                    float* __restrict__ out, int B, int rows) {
  __shared__ __align__(16) float Ws[T_PAD * SEQ];    // [t][s]  16x80  (5 KB)
  __shared__ __align__(16) float Xs[2][16 * DIM];    // 2 x 16 K-rows  (32 KB)

  const int b = blockIdx.x;
  const int tid = threadIdx.x;   // 0..127
  const int wave = tid >> 5;     // 0..3  -> N tiles 4*wave .. 4*wave+3
  const int lane = tid & 31;
  const int lm = lane & 15;
  const int lh = lane >> 4;

  // Stage weight tile Ws[t][s] = w[t][b*80+s] (rows t>=11 are zero in w).
  for (int idx = tid; idx < T_PAD * SEQ; idx += 128) {
    const int t = idx / SEQ;
    const int s = idx - t * SEQ;
    Ws[idx] = w[(size_t)t * rows + b * SEQ + s];
  }

  const float* xb = x + (size_t)b * SEQ * DIM;
  stage_chunk(xb, Xs[0], tid);          // chunk 0
  wait_async_copies();
  __syncthreads();

  v8f acc[4] = {};

  const int KCHUNKS = SEQ / 16;         // 5
  for (int c = 0; c < KCHUNKS; ++c) {
    const int buf = c & 1;
    if (c + 1 < KCHUNKS)                // overlap next copy with compute
      stage_chunk(xb + (size_t)(c + 1) * 16 * DIM, Xs[(c + 1) & 1], tid);

    const float* wsrow = Ws + lm * SEQ + c * 16 + 2 * lh;  // 8B aligned
    #pragma unroll
    for (int kk = 0; kk < 4; ++kk) {
      v2f a = *(const v2f*)(wsrow + kk * 4);
      const float* xrow = Xs[buf] + (kk * 4 + 2 * lh) * DIM;
      #pragma unroll
      for (int i = 0; i < 4; ++i) {
        const int d0 = (wave * 4 + i) * 16;
        v2f bf;
        bf.x = xrow[d0 + lm];
        bf.y = xrow[DIM + d0 + lm];
        acc[i] = __builtin_amdgcn_wmma_f32_16x16x4_f32(false, a, false, bf,
                                                       (short)0, acc[i],
                                                       false, false);
      }
    }
    wait_async_copies();                // next chunk fully in LDS
    __syncthreads();                    // all waves done reading old buffer
  }

  #pragma unroll
  for (int i = 0; i < 4; ++i) {
    const int nt = wave * 4 + i;
    #pragma unroll
    for (int v = 0; v < 8; ++v) {
      const int t = v + 8 * lh;
      if (t < T_REAL)
        out[((size_t)t * B + b) * DIM + nt * 16 + lm] = acc[i][v];
    }
  }
}

extern "C" void kernel_launch(void* const* d_in, const int* in_sizes, int n_in,
                              void* d_out, int out_size, void* d_ws, size_t ws_size,
                              hipStream_t stream) {
  const float* x    = (const float*)d_in[0];   // (B, 80, 256) f32
  const float* kern = (const float*)d_in[1];   // (256, 11) f32
  float* out = (float*)d_out;                  // (11, B, 256) f32

  const int B = in_sizes[0] / (SEQ * DIM);     // 256
  const int rows = B * SEQ;                    // 20480

  // Workspace: scores (rows x 16 f32) then weights (16 x rows f32) ~ 2.5 MB.
  float* sc = (float*)d_ws;
  float* w  = sc + (size_t)rows * T_PAD;

  // 1280 M-tiles, 4 waves (128 threads) per block.
  scores_kernel<<<dim3(rows / 64), dim3(128), 0, stream>>>(x, kern, sc);
  softmax_kernel<<<dim3(T_PAD), dim3(B), 0, stream>>>(sc, w, B, rows);
  weighted_sum_kernel<<<dim3(B), dim3(128), 0, stream>>>(x, w, out, B, rows);
}